// decoder_model_89481348645482
// MI455X (gfx1250) — compile-verified
//
#include <hip/hip_runtime.h>
#include <hip/hip_bf16.h>
#include <math.h>

typedef __attribute__((ext_vector_type(2))) float v2f;
typedef __attribute__((ext_vector_type(8))) float v8f;

#define BDIM  64
#define HDIM  512
#define D1DIM 11264   // S*E + 2*E = 20*512 + 1024
#define SEDIM 10240   // S*E
#define NGATE 2048    // 4*H
#define VDIM  32000

// ---------------------------------------------------------------------------
// Build x = [flat_enc | word_embedding | persona[speaker]]  -> (64, 11264)
// ---------------------------------------------------------------------------
__global__ void gather_x_kernel(const float* __restrict__ enc,
                                const float* __restrict__ wemb,
                                const float* __restrict__ persona,
                                const int*   __restrict__ speaker,
                                float* __restrict__ X) {
  const int b = blockIdx.x;
  const int t = threadIdx.x;                 // 256 threads
  float* xr = X + b * D1DIM;
  const float* er = enc + b * SEDIM;
  for (int i = t; i < SEDIM; i += 256) xr[i] = er[i];
  const float* wr = wemb + b * HDIM;
  for (int i = t; i < HDIM; i += 256) xr[SEDIM + i] = wr[i];
  const float* pr = persona + speaker[b] * HDIM;
  for (int i = t; i < HDIM; i += 256) xr[SEDIM + HDIM + i] = pr[i];
}

// ---------------------------------------------------------------------------
// Z(64 x N) = / += A(64 x K) @ B(K x N) using V_WMMA_F32_16X16X4_F32.
// One wave per 16-wide N strip computes all 64 rows (weights read exactly
// once, non-temporal); split-K across gridDim.y. accumulate!=0 -> float
// atomics into zeroed Z; accumulate==0 (single writer) -> plain NT stores.
// Block = 32 threads (one wave, EXEC all ones as WMMA requires).
// All indexing is int32: every buffer is < 2^31 elements, which lets codegen
// use saddr + 32-bit voffset instead of per-lane 64-bit address math.
// K-loop unrolled x2: 8 WMMAs + 12 loads per iteration for load/MMA overlap.
// ---------------------------------------------------------------------------
__global__ __launch_bounds__(32)
void gemm_wmma(const float* __restrict__ A, int lda,
               const float* __restrict__ B, int ldb,
               float* __restrict__ Z, int ldz,
               int K, int kChunk, int accumulate) {
  const int n0   = blockIdx.x * 16;
  const int k0   = blockIdx.y * kChunk;
  const int kEnd = (k0 + kChunk < K) ? (k0 + kChunk) : K;
  const int lane = (int)threadIdx.x;
  const int half = lane >> 4;      // 0: K pair {0,1}, 1: K pair {2,3}
  const int l15  = lane & 15;

  v8f acc0 = {}, acc1 = {}, acc2 = {}, acc3 = {};

  // Incrementally updated 32-bit offsets (cheap v_add_nc_u32 per step).
  int aoff = l15 * lda + k0 + 2 * half;          // A[l15][k0 + 2*half]
  int boff = (k0 + 2 * half) * ldb + n0 + l15;   // B[k0 + 2*half][n0 + l15]
  const int a16   = 16 * lda;                    // next M-tile
  const int bstep = 4 * ldb;                     // next K group of 4

  for (int k = k0; k < kEnd; k += 8) {
    // ---- K group 0 (k .. k+3) ----
    v2f bfa;
    bfa.x = __builtin_nontemporal_load(B + boff);
    bfa.y = __builtin_nontemporal_load(B + boff + ldb);
    v2f af0a = *(const v2f*)(A + aoff);
    v2f af1a = *(const v2f*)(A + aoff + a16);
    v2f af2a = *(const v2f*)(A + aoff + 2 * a16);
    v2f af3a = *(const v2f*)(A + aoff + 3 * a16);
    // ---- K group 1 (k+4 .. k+7) ----
    v2f bfb;
    bfb.x = __builtin_nontemporal_load(B + boff + bstep);
    bfb.y = __builtin_nontemporal_load(B + boff + bstep + ldb);
    v2f af0b = *(const v2f*)(A + aoff + 4);
    v2f af1b = *(const v2f*)(A + aoff + 4 + a16);
    v2f af2b = *(const v2f*)(A + aoff + 4 + 2 * a16);
    v2f af3b = *(const v2f*)(A + aoff + 4 + 3 * a16);

    acc0 = __builtin_amdgcn_wmma_f32_16x16x4_f32(false, af0a, false, bfa, (short)0, acc0, false, false);
    acc1 = __builtin_amdgcn_wmma_f32_16x16x4_f32(false, af1a, false, bfa, (short)0, acc1, false, false);
    acc2 = __builtin_amdgcn_wmma_f32_16x16x4_f32(false, af2a, false, bfa, (short)0, acc2, false, false);
    acc3 = __builtin_amdgcn_wmma_f32_16x16x4_f32(false, af3a, false, bfa, (short)0, acc3, false, false);
    acc0 = __builtin_amdgcn_wmma_f32_16x16x4_f32(false, af0b, false, bfb, (short)0, acc0, false, false);
    acc1 = __builtin_amdgcn_wmma_f32_16x16x4_f32(false, af1b, false, bfb, (short)0, acc1, false, false);
    acc2 = __builtin_amdgcn_wmma_f32_16x16x4_f32(false, af2b, false, bfb, (short)0, acc2, false, false);
    acc3 = __builtin_amdgcn_wmma_f32_16x16x4_f32(false, af3b, false, bfb, (short)0, acc3, false, false);

    aoff += 8;
    boff += 2 * bstep;
  }

  // C/D layout: VGPR v, lanes 0-15 -> M=v, lanes 16-31 -> M=v+8, N=l15.
  v8f accs[4] = {acc0, acc1, acc2, acc3};
  if (accumulate) {
#pragma unroll
    for (int mt = 0; mt < 4; ++mt) {
      int zoff = (mt * 16 + 8 * half) * ldz + n0 + l15;
#pragma unroll
      for (int v = 0; v < 8; ++v)
        atomicAdd(Z + zoff + v * ldz, accs[mt][v]);
    }
  } else {
#pragma unroll
    for (int mt = 0; mt < 4; ++mt) {
      int zoff = (mt * 16 + 8 * half) * ldz + n0 + l15;
#pragma unroll
      for (int v = 0; v < 8; ++v)
        __builtin_nontemporal_store(accs[mt][v], Z + zoff + v * ldz);
    }
  }
}

// ---------------------------------------------------------------------------
// LSTM gate math (Keras order i,f,g,o; activation relu, recurrent sigmoid).
// z (64 x 2048, no bias yet) + bias -> h_new, c_new. Optional mirror copies.
// ---------------------------------------------------------------------------
__global__ void lstm_pointwise(const float* __restrict__ z,
                               const float* __restrict__ bias,
                               const float* __restrict__ c_in,
                               float* __restrict__ h_out,
                               float* __restrict__ c_out,
                               float* __restrict__ h_copy,
                               float* __restrict__ c_copy) {
  const int idx = blockIdx.x * blockDim.x + threadIdx.x;
  if (idx >= BDIM * HDIM) return;
  const int b = idx >> 9;            // / 512
  const int n = idx & (HDIM - 1);
  const float* zr = z + b * NGATE;
  const float zi = zr[n]            + bias[n];
  const float zf = zr[HDIM + n]     + bias[HDIM + n];
  const float zg = zr[2 * HDIM + n] + bias[2 * HDIM + n];
  const float zo = zr[3 * HDIM + n] + bias[3 * HDIM + n];
  const float si = 1.0f / (1.0f + __expf(-zi));
  const float sf = 1.0f / (1.0f + __expf(-zf));
  const float so = 1.0f / (1.0f + __expf(-zo));
  const float g  = fmaxf(zg, 0.0f);
  const float cn = sf * c_in[idx] + si * g;
  const float hn = so * fmaxf(cn, 0.0f);
  c_out[idx] = cn;
  h_out[idx] = hn;
  if (h_copy) { h_copy[idx] = hn; c_copy[idx] = cn; }
}

// ---------------------------------------------------------------------------
// In-place row softmax of logits(64 x 32000) with bias added on the fly.
// ---------------------------------------------------------------------------
__global__ void softmax_rows(float* __restrict__ logits,
                             const float* __restrict__ bias) {
  __shared__ float red[256];
  const int row = blockIdx.x;
  const int t   = threadIdx.x;
  float* lr = logits + row * VDIM;

  float mx = -INFINITY;
  for (int i = t; i < VDIM; i += 256) mx = fmaxf(mx, lr[i] + bias[i]);
  red[t] = mx; __syncthreads();
  for (int s = 128; s > 0; s >>= 1) {
    if (t < s) red[t] = fmaxf(red[t], red[t + s]);
    __syncthreads();
  }
  mx = red[0]; __syncthreads();

  float sum = 0.0f;
  for (int i = t; i < VDIM; i += 256) sum += __expf(lr[i] + bias[i] - mx);
  red[t] = sum; __syncthreads();
  for (int s = 128; s > 0; s >>= 1) {
    if (t < s) red[t] += red[t + s];
    __syncthreads();
  }
  const float inv = 1.0f / red[0];

  for (int i = t; i < VDIM; i += 256) lr[i] = __expf(lr[i] + bias[i] - mx) * inv;
}

// ---------------------------------------------------------------------------
extern "C" void kernel_launch(void* const* d_in, const int* in_sizes, int n_in,
                              void* d_out, int out_size, void* d_ws, size_t ws_size,
                              hipStream_t stream) {
  const float* enc     = (const float*)d_in[0];
  const float* wemb    = (const float*)d_in[1];
  const float* h0      = (const float*)d_in[2];
  const float* c0      = (const float*)d_in[3];
  const float* persona = (const float*)d_in[4];
  const float* W[4] = {(const float*)d_in[5],  (const float*)d_in[8],
                       (const float*)d_in[11], (const float*)d_in[14]};
  const float* U[4] = {(const float*)d_in[6],  (const float*)d_in[9],
                       (const float*)d_in[12], (const float*)d_in[15]};
  const float* bb[4] = {(const float*)d_in[7], (const float*)d_in[10],
                        (const float*)d_in[13], (const float*)d_in[16]};
  const float* Wd = (const float*)d_in[17];
  const float* bd = (const float*)d_in[18];
  const int* speaker = (const int*)d_in[19];

  // Workspace layout (floats)
  float* ws = (float*)d_ws;
  float* X = ws;                                  // 64*11264 = 720896
  float* z = X + (size_t)BDIM * D1DIM;            // 64*2048  = 131072
  float* h = z + (size_t)BDIM * NGATE;            // 64*512
  float* c = h + (size_t)BDIM * HDIM;             // 64*512

  // Output layout
  float* probs  = (float*)d_out;                  // 64*32000
  float* h4_out = probs + (size_t)BDIM * VDIM;    // 64*512
  float* c4_out = h4_out + (size_t)BDIM * HDIM;   // 64*512

  // 1) Gather input features
  gather_x_kernel<<<BDIM, 256, 0, stream>>>(enc, wemb, persona, speaker, X);

  // 2) Layer 1: z = X@W1 + h0@U1 ; then gates
  hipMemsetAsync(z, 0, sizeof(float) * BDIM * NGATE, stream);
  gemm_wmma<<<dim3(NGATE / 16, 16), 32, 0, stream>>>(X, D1DIM, W[0], NGATE,
                                                     z, NGATE, D1DIM, 704, 1);
  gemm_wmma<<<dim3(NGATE / 16, 4), 32, 0, stream>>>(h0, HDIM, U[0], NGATE,
                                                    z, NGATE, HDIM, 128, 1);
  lstm_pointwise<<<(BDIM * HDIM + 255) / 256, 256, 0, stream>>>(
      z, bb[0], c0, h, c, nullptr, nullptr);

  // 3) Layers 2..4: z = h@(W_l) + h@(U_l) ; state in-place in ws
  for (int l = 1; l < 4; ++l) {
    hipMemsetAsync(z, 0, sizeof(float) * BDIM * NGATE, stream);
    gemm_wmma<<<dim3(NGATE / 16, 4), 32, 0, stream>>>(h, HDIM, W[l], NGATE,
                                                      z, NGATE, HDIM, 128, 1);
    gemm_wmma<<<dim3(NGATE / 16, 4), 32, 0, stream>>>(h, HDIM, U[l], NGATE,
                                                      z, NGATE, HDIM, 128, 1);
    const bool last = (l == 3);
    lstm_pointwise<<<(BDIM * HDIM + 255) / 256, 256, 0, stream>>>(
        z, bb[l], c, h, c,
        last ? h4_out : nullptr, last ? c4_out : nullptr);
  }

  // 4) Decoder: single K chunk -> sole writer -> direct NT stores, no memset,
  //    no atomics. 2000 waves is ample occupancy for a 65 MB weight stream.
  gemm_wmma<<<dim3(VDIM / 16, 1), 32, 0, stream>>>(h, HDIM, Wd, VDIM,
                                                   probs, VDIM, HDIM, HDIM, 0);
  softmax_rows<<<BDIM, 256, 0, stream>>>(probs, bd);
}